// PositionalEmbeddingLayer_40845138985515
// MI455X (gfx1250) — compile-verified
//
#include <hip/hip_runtime.h>
#include <hip/hip_bf16.h>
#include <stdint.h>

// Segment-relative normalized positions prepended to x.
//   out[row, 0]   = (row - seg_start + 1) / L
//   out[row, 1+c] = x[row, c],  c in [0,128)
// Pure streaming op: ~1.03 GB HBM traffic -> ~44us floor at 23.3 TB/s.
// Bulk payload flows global --(async DMA)--> LDS --(async DMA)--> global on
// the CDNA5 ASYNCcnt path; VGPRs/VALU stay nearly idle. 4 rows (2KB) are kept
// in flight per wave between ASYNCcnt drains for memory-level parallelism.

#define D_FEAT 128
#define ROW_IN_BYTES  (D_FEAT * 4)        // 512
#define ROW_OUT_BYTES ((D_FEAT + 1) * 4)  // 516
#define WAVES_PER_BLOCK 8
#define THREADS_PER_BLOCK (WAVES_PER_BLOCK * 32)
#define ROWS_PER_WAVE 4                   // rows per wave per drain cycle
#define ROWS_PER_ITER (WAVES_PER_BLOCK * ROWS_PER_WAVE)

// ---------------------------------------------------------------------------
// Kernel 1: exclusive prefix sum of lengths -> offs (single block, 256 thr).
// Ragged-safe: handles any lengths that sum to the total row count.
// ---------------------------------------------------------------------------
__global__ void seg_scan_offsets(const int* __restrict__ lengths, int n_seg,
                                 int* __restrict__ offs) {
    __shared__ int partial[256];
    const int tid   = threadIdx.x;
    const int chunk = (n_seg + 255) / 256;
    const int base  = tid * chunk;

    int s = 0;
    for (int i = 0; i < chunk; ++i) {
        int j = base + i;
        if (j < n_seg) s += lengths[j];
    }
    partial[tid] = s;
    __syncthreads();

    // Hillis-Steele inclusive scan over the 256 chunk sums.
    for (int off = 1; off < 256; off <<= 1) {
        int v = (tid >= off) ? partial[tid - off] : 0;
        __syncthreads();
        partial[tid] += v;
        __syncthreads();
    }

    int run = (tid == 0) ? 0 : partial[tid - 1];  // exclusive prefix of chunk
    for (int i = 0; i < chunk; ++i) {
        int j = base + i;
        if (j < n_seg) {
            offs[j] = run;
            run += lengths[j];
        }
    }
}

// ---------------------------------------------------------------------------
// Kernel 2: one workgroup per segment; each wave32 streams 4 x 512B rows per
// drain cycle through LDS staging slots via CDNA5 async global<->LDS DMA.
// ---------------------------------------------------------------------------
__global__ void __launch_bounds__(THREADS_PER_BLOCK)
seg_pos_concat(const float* __restrict__ x,
               const int* __restrict__ lengths,
               const int* __restrict__ offs,
               float* __restrict__ out) {
    // 4 row slots of 512B per wave -> 16KB static LDS per block.
    __shared__ __align__(16)
        unsigned char stage[WAVES_PER_BLOCK * ROWS_PER_WAVE * ROW_IN_BYTES];

    const int seg   = blockIdx.x;
    const int L     = lengths[seg];
    const int start = offs[seg];
    const int lane  = threadIdx.x & 31;
    const int wave  = threadIdx.x >> 5;
    const float invL = 1.0f / (float)L;

    // LDS byte offset of this wave's staging area (low 32 bits of an LDS
    // pointer are the wave-relative LDS address the async DMA ops consume).
    const unsigned lds_wave =
        (unsigned)(uintptr_t)(stage + wave * (ROWS_PER_WAVE * ROW_IN_BYTES));
    const unsigned lane16 = (unsigned)(lane * 16);

    for (int r0 = wave * ROWS_PER_WAVE; r0 < L; r0 += ROWS_PER_ITER) {
        // LDS slots must be free: previous cycle's async stores drained.
        asm volatile("s_wait_asynccnt 0x0" ::: "memory");

        // ---- Phase 1: batch async loads, 512B per row, 16B/lane, aligned.
        #pragma unroll
        for (int k = 0; k < ROWS_PER_WAVE; ++k) {
            const int r = r0 + k;                       // wave-uniform guard
            if (r < L) {
                const unsigned row = (unsigned)(start + r);   // < 2^20
                const unsigned gin = row * (unsigned)ROW_IN_BYTES + lane16;
                const unsigned lds = lds_wave + (unsigned)(k * ROW_IN_BYTES) + lane16;
                asm volatile("global_load_async_to_lds_b128 %0, %1, %2"
                             :
                             : "v"(lds), "v"(gin), "s"(x)
                             : "memory");
            }
        }
        asm volatile("s_wait_asynccnt 0x0" ::: "memory");   // loads landed

        // ---- Phase 2: batch async stores. Output payload starts at
        // row*516+4 (4B-aligned only -> b32): 4 x (32 lanes x 4B) per row.
        #pragma unroll
        for (int k = 0; k < ROWS_PER_WAVE; ++k) {
            const int r = r0 + k;                       // wave-uniform guard
            if (r < L) {
                const unsigned row  = (unsigned)(start + r);
                const unsigned gout = row * (unsigned)ROW_OUT_BYTES + 4u;
                const unsigned lrow = lds_wave + (unsigned)(k * ROW_IN_BYTES);
                #pragma unroll
                for (int i = 0; i < 4; ++i) {
                    const unsigned d   = (unsigned)(lane + 32 * i); // dword 0..127
                    const unsigned ldw = lrow + d * 4u;
                    const unsigned gdw = gout + d * 4u;
                    asm volatile("global_store_async_from_lds_b32 %0, %1, %2"
                                 :
                                 : "v"(gdw), "v"(ldw), "s"(out)
                                 : "memory");
                }
            }
        }

        // ---- Position column: lanes 0..3 each write one row's pos dword
        // (normal STOREcnt store, disjoint from the async payload bytes).
        if (lane < ROWS_PER_WAVE) {
            const int r = r0 + lane;
            if (r < L) {
                out[(size_t)(start + r) * (D_FEAT + 1)] = (float)(r + 1) * invL;
            }
        }
    }

    // Drain outstanding async stores before wave end.
    asm volatile("s_wait_asynccnt 0x0" ::: "memory");
}

// ---------------------------------------------------------------------------
// Launch: inputs are (x: float32 [total*128], lengths: int32 [n_seg]).
// d_ws holds the n_seg exclusive offsets.
// ---------------------------------------------------------------------------
extern "C" void kernel_launch(void* const* d_in, const int* in_sizes, int n_in,
                              void* d_out, int out_size, void* d_ws, size_t ws_size,
                              hipStream_t stream) {
    (void)n_in; (void)out_size; (void)ws_size;

    const float* x      = (const float*)d_in[0];
    const int* lengths  = (const int*)d_in[1];
    const int n_seg     = in_sizes[1];
    int* offs           = (int*)d_ws;
    float* out          = (float*)d_out;

    seg_scan_offsets<<<1, 256, 0, stream>>>(lengths, n_seg, offs);
    seg_pos_concat<<<n_seg, THREADS_PER_BLOCK, 0, stream>>>(x, lengths, offs, out);
}